// NetGCN_74801150427782
// MI455X (gfx1250) — compile-verified
//
#include <hip/hip_runtime.h>

typedef __attribute__((ext_vector_type(2))) float v2f;
typedef __attribute__((ext_vector_type(8))) float v8f;

#define N_NODES   100000
#define N_EDGES   3200000
#define NFEAT     128
#define DIM       10
#define DIMP      16      // padded feature dim for WMMA tiles / vector loads
#define NGRAPH    64

// ---------------------------------------------------------------------------
// K1: h1[N,16] = features[N,128] @ W1pad[128,16]   (W1pad cols 10..15 = 0)
// One wave per 16-row tile, V_WMMA_F32_16X16X4_F32, 32 k-steps (K=128).
// 6250 tiles * 16 rows = exactly 100000 -> no row tail.
// ---------------------------------------------------------------------------
__global__ __launch_bounds__(256) void gemm1_wmma(const float* __restrict__ F,
                                                  const float* __restrict__ W1,
                                                  float* __restrict__ h1) {
    __shared__ float Wlds[NFEAT * DIMP];  // 8 KB, shared by all 8 waves
    const int tid = threadIdx.x;
    for (int i = tid; i < NFEAT * DIMP; i += 256) {
        const int k = i >> 4, n = i & 15;
        Wlds[i] = (n < DIM) ? W1[k * DIM + n] : 0.0f;
    }
    __syncthreads();

    const int wave = tid >> 5;
    const int lane = tid & 31;
    const int tile = blockIdx.x * 8 + wave;
    if (tile * 16 >= N_NODES) return;     // wave-uniform exit (EXEC all-ones inside)

    const int rowbase = tile * 16;
    const int m    = lane & 15;           // A: row within tile / B,C: column
    const int koff = (lane >> 4) << 1;    // 0 for lanes 0-15, 2 for lanes 16-31
    const float* frow = F + (size_t)(rowbase + m) * NFEAT + koff;

    v8f c = {0.f, 0.f, 0.f, 0.f, 0.f, 0.f, 0.f, 0.f};
    #pragma unroll 4
    for (int kk = 0; kk < NFEAT; kk += 4) {
        // A fragment: 16x4 f32 tile; v0/v1 = K = koff, koff+1 (per ISA layout)
        v2f a = *(const v2f*)(frow + kk);
        // B fragment: 4x16 f32 tile, N striped across lanes
        const int k0 = kk + koff;
        v2f b;
        b.x = Wlds[k0 * DIMP + m];
        b.y = Wlds[(k0 + 1) * DIMP + m];
        c = __builtin_amdgcn_wmma_f32_16x16x4_f32(false, a, false, b,
                                                  (short)0, c, false, false);
    }

    // C/D layout: VGPR r holds M=r (lanes 0-15) and M=r+8 (lanes 16-31), N=lane&15
    const int rowoff = (lane >> 4) * 8;
    float* out = h1 + (size_t)(rowbase + rowoff) * DIMP + m;
    #pragma unroll
    for (int r = 0; r < 8; ++r) out[(size_t)r * DIMP] = c[r];
}

// ---------------------------------------------------------------------------
// Edge scatter-add: hout[dst[e], d] += (relu?)(hin[src[e], d]),  d in [0,10)
// 16 lanes per edge (contiguous loads, L2-resident f32 atomics).
// ---------------------------------------------------------------------------
template <bool RELU>
__global__ __launch_bounds__(256) void edge_scatter(const float* __restrict__ hin,
                                                    const int* __restrict__ src,
                                                    const int* __restrict__ dst,
                                                    float* __restrict__ hout) {
    const long long total  = (long long)N_EDGES * 16;
    const long long stride = (long long)gridDim.x * blockDim.x;
    for (long long idx = (long long)blockIdx.x * blockDim.x + threadIdx.x;
         idx < total; idx += stride) {
        const int d = (int)(idx & 15);
        if (d >= DIM) continue;
        const int e = (int)(idx >> 4);
        const int s = src[e];
        const int t = dst[e];
        float v = hin[(size_t)s * DIMP + d];
        if (RELU) v = fmaxf(v, 0.0f);
        atomicAdd(&hout[(size_t)t * DIMP + d], v);
    }
}

// ---------------------------------------------------------------------------
// Per-graph sums of h3 plus node counts (graph_ids sorted, 64 graphs).
// ---------------------------------------------------------------------------
__global__ __launch_bounds__(256) void pool_sum(const float* __restrict__ h3,
                                                const int* __restrict__ gids,
                                                float* __restrict__ S,
                                                float* __restrict__ cnt) {
    const long long total  = (long long)N_NODES * 16;
    const long long stride = (long long)gridDim.x * blockDim.x;
    for (long long idx = (long long)blockIdx.x * blockDim.x + threadIdx.x;
         idx < total; idx += stride) {
        const int d    = (int)(idx & 15);
        const int node = (int)(idx >> 4);
        const int g    = gids[node];
        if (d < DIM) {
            atomicAdd(&S[g * DIM + d], h3[(size_t)node * DIMP + d]);
        } else if (d == DIM) {
            atomicAdd(&cnt[g], 1.0f);
        }
    }
}

// ---------------------------------------------------------------------------
// Head: out[g] = sigmoid( (S[g]/cnt[g]) . (W2 @ W3) )
// (mean_nodes is linear, so GEMM2 + pooling + W3 fold into one dot product.)
// ---------------------------------------------------------------------------
__global__ void head_kernel(const float* __restrict__ S,
                            const float* __restrict__ cnt,
                            const float* __restrict__ W2,
                            const float* __restrict__ W3,
                            float* __restrict__ out) {
    const int g = threadIdx.x;
    if (g >= NGRAPH) return;
    float wf[DIM];
    #pragma unroll
    for (int d = 0; d < DIM; ++d) {
        float acc = 0.0f;
        #pragma unroll
        for (int j = 0; j < DIM; ++j) acc += W2[d * DIM + j] * W3[j];
        wf[d] = acc;
    }
    const float c = fmaxf(cnt[g], 1.0f);
    float val = 0.0f;
    #pragma unroll
    for (int d = 0; d < DIM; ++d) val += (S[g * DIM + d] / c) * wf[d];
    out[g] = 1.0f / (1.0f + __expf(-val));
}

// ---------------------------------------------------------------------------
// Workspace zeroing (graph-capture safe; re-run every launch -> deterministic)
// ---------------------------------------------------------------------------
__global__ __launch_bounds__(256) void zero_f32(float* __restrict__ p, long long n) {
    const long long stride = (long long)gridDim.x * blockDim.x;
    for (long long i = (long long)blockIdx.x * blockDim.x + threadIdx.x;
         i < n; i += stride) p[i] = 0.0f;
}

// ---------------------------------------------------------------------------
// Workspace layout (floats):
//   h1 : N_NODES*16            = 1,600,000
//   h2 : N_NODES*16            = 1,600,000
//   h3 : N_NODES*16            = 1,600,000
//   S  : NGRAPH*DIM            = 640
//   cnt: NGRAPH                = 64
// total ~19.2 MB
// ---------------------------------------------------------------------------
extern "C" void kernel_launch(void* const* d_in, const int* in_sizes, int n_in,
                              void* d_out, int out_size, void* d_ws, size_t ws_size,
                              hipStream_t stream) {
    const float* features = (const float*)d_in[0];
    const float* W1       = (const float*)d_in[1];
    const float* W2       = (const float*)d_in[2];
    const float* W3       = (const float*)d_in[3];
    const int*   src      = (const int*)d_in[4];
    const int*   dst      = (const int*)d_in[5];
    const int*   gids     = (const int*)d_in[6];
    float*       out      = (float*)d_out;

    float* ws  = (float*)d_ws;
    float* h1  = ws;
    float* h2  = h1 + (size_t)N_NODES * DIMP;
    float* h3  = h2 + (size_t)N_NODES * DIMP;
    float* S   = h3 + (size_t)N_NODES * DIMP;
    float* cnt = S + NGRAPH * DIM;

    // zero everything the atomics accumulate into (h2, h3, S, cnt contiguous)
    const long long nzero = (long long)2 * N_NODES * DIMP + NGRAPH * DIM + NGRAPH;
    zero_f32<<<2048, 256, 0, stream>>>(h2, nzero);

    // conv1 weight transform via WMMA
    gemm1_wmma<<<(N_NODES / 16 + 7) / 8, 256, 0, stream>>>(features, W1, h1);

    // conv1 aggregation
    edge_scatter<false><<<2048, 256, 0, stream>>>(h1, src, dst, h2);

    // relu fused into conv2 aggregation read
    edge_scatter<true><<<2048, 256, 0, stream>>>(h2, src, dst, h3);

    // per-graph sums + counts
    pool_sum<<<2048, 256, 0, stream>>>(h3, gids, S, cnt);

    // fold GEMM2 + mean + W3 + sigmoid
    head_kernel<<<1, 64, 0, stream>>>(S, cnt, W2, W3, out);
}